// DenoisingNet_42932493091134
// MI455X (gfx1250) — compile-verified
//
#include <hip/hip_runtime.h>
#include <hip/hip_bf16.h>

// ---------------- problem constants ----------------
#define N_USER   100000
#define N_ITEM   50000
#define NN       150000          // N_USER + N_ITEM
#define EE       1000000
#define DD       64
#define BB       4096
#define GAMMA    (-0.45f)
#define ZETA     (1.05f)
#define LAMBDA0  (0.0001f)
#define REG_W    (0.0001f)
#define LT_CONST (-0.84729786f)  // log(-GAMMA/ZETA) = log(0.45/1.05)

typedef __attribute__((ext_vector_type(2))) float v2f;
typedef __attribute__((ext_vector_type(8))) float v8f;

__device__ __forceinline__ void atomicAddF(float* p, float v) {
    __hip_atomic_fetch_add(p, v, __ATOMIC_RELAXED, __HIP_MEMORY_SCOPE_AGENT);
}

__device__ __forceinline__ float waveReduce(float v) {
    #pragma unroll
    for (int o = 16; o > 0; o >>= 1) v += __shfl_xor(v, o, 32);
    return v;
}

// ---------------- zero helpers ----------------
__global__ void zero_f_kernel(float* __restrict__ p, long n) {
    long i = (long)blockIdx.x * blockDim.x + threadIdx.x;
    if (i < n) p[i] = 0.f;
}
__global__ void zero_i_kernel(int* __restrict__ p, long n) {
    long i = (long)blockIdx.x * blockDim.x + threadIdx.x;
    if (i < n) p[i] = 0;
}

// ---------------------------------------------------------------------------
// CSR build (row/col identical for both layers -> build once per launch).
// ---------------------------------------------------------------------------
__global__ __launch_bounds__(256)
void count_kernel(const int* __restrict__ row, int* __restrict__ cnt, int E) {
    const int e = blockIdx.x * blockDim.x + threadIdx.x;
    if (e < E) atomicAdd(&cnt[row[e]], 1);
}

// single-workgroup chunked Hillis-Steele exclusive scan (n = 150K, trivial time)
__global__ __launch_bounds__(1024)
void scan_kernel(const int* __restrict__ cnt, int* __restrict__ off, int n) {
    __shared__ int buf[1024];
    __shared__ int carry;
    if (threadIdx.x == 0) carry = 0;
    __syncthreads();
    for (int base = 0; base < n; base += 1024) {
        const int i = base + (int)threadIdx.x;
        const int v = (i < n) ? cnt[i] : 0;
        buf[threadIdx.x] = v;
        __syncthreads();
        #pragma unroll
        for (int s = 1; s < 1024; s <<= 1) {
            const int t = (threadIdx.x >= (unsigned)s) ? buf[threadIdx.x - s] : 0;
            __syncthreads();
            buf[threadIdx.x] += t;
            __syncthreads();
        }
        const int incl = buf[threadIdx.x];
        if (i < n) off[i] = carry + (incl - v);
        __syncthreads();
        if (threadIdx.x == 1023) carry += incl;
        __syncthreads();
    }
    if (threadIdx.x == 0) off[n] = carry;   // == E
}

__global__ __launch_bounds__(256)
void copy_i_kernel(const int* __restrict__ src, int* __restrict__ dst, int n) {
    const int i = blockIdx.x * blockDim.x + threadIdx.x;
    if (i < n) dst[i] = src[i];
}

__global__ __launch_bounds__(256)
void scatter_kernel(const int* __restrict__ row, const int* __restrict__ col,
                    int* __restrict__ cursor, int* __restrict__ csr_col,
                    int* __restrict__ csr_eid, int E) {
    const int e = blockIdx.x * blockDim.x + threadIdx.x;
    if (e < E) {
        const int p = atomicAdd(&cursor[row[e]], 1);
        csr_col[p] = col[e];
        csr_eid[p] = e;
    }
}

// ---------------------------------------------------------------------------
// Node-level attention precompute with fp32 WMMA:
//   a_nb[n]   = relu(x[n] @ Wn + bn) . Wa[0:64]
//   a_self[n] = relu(x[n] @ Ws + bs) . Wa[64:128]
// One wave per 16-node tile; V_WMMA_F32_16X16X4_F32, K accumulated over 16
// steps, 4 n-tiles, both matrices share the A fragments.
// ---------------------------------------------------------------------------
#define TPB_TILES 4
__global__ __launch_bounds__(128)
void node_attn_kernel(const float* __restrict__ x,
                      const float* __restrict__ Wn, const float* __restrict__ bn,
                      const float* __restrict__ Ws, const float* __restrict__ bs,
                      const float* __restrict__ Wa,      // 128
                      int n_tiles,
                      float* __restrict__ a_nb, float* __restrict__ a_self)
{
    __shared__ float lds[TPB_TILES][2][16][DD + 1];   // +1 pad: no bank conflicts

    const int wave  = threadIdx.x >> 5;
    const int lane  = threadIdx.x & 31;
    const int m     = lane & 15;
    const int khalf = lane >> 4;

    const int tile   = blockIdx.x * TPB_TILES + wave;
    const int tile_c = tile < n_tiles ? tile : (n_tiles - 1);   // clamp: keep barrier safe
    const float* xt  = x + (size_t)tile_c * 16 * DD;

    v8f cn[4] = {v8f{0}, v8f{0}, v8f{0}, v8f{0}};
    v8f cs[4] = {v8f{0}, v8f{0}, v8f{0}, v8f{0}};

    #pragma unroll 2
    for (int kk = 0; kk < 16; ++kk) {
        const int k0 = 4 * kk + 2 * khalf;
        v2f a;
        a.x = xt[m * DD + k0];
        a.y = xt[m * DD + k0 + 1];
        #pragma unroll
        for (int j = 0; j < 4; ++j) {
            const int col = 16 * j + m;
            v2f bnf, bsf;
            bnf.x = Wn[k0 * DD + col];
            bnf.y = Wn[(k0 + 1) * DD + col];
            bsf.x = Ws[k0 * DD + col];
            bsf.y = Ws[(k0 + 1) * DD + col];
            cn[j] = __builtin_amdgcn_wmma_f32_16x16x4_f32(false, a, false, bnf,
                                                          (short)0, cn[j], false, false);
            cs[j] = __builtin_amdgcn_wmma_f32_16x16x4_f32(false, a, false, bsf,
                                                          (short)0, cs[j], false, false);
        }
    }

    // bias + relu, stage H tiles to LDS (C layout: vgpr v -> row v / v+8)
    #pragma unroll
    for (int j = 0; j < 4; ++j) {
        const int col = 16 * j + m;
        const float bnb = bn[col], bsl = bs[col];
        #pragma unroll
        for (int v = 0; v < 8; ++v) {
            const int r = v + 8 * khalf;
            float hn = cn[j][v] + bnb; hn = fmaxf(hn, 0.f);
            float hs = cs[j][v] + bsl; hs = fmaxf(hs, 0.f);
            lds[wave][0][r][col] = hn;
            lds[wave][1][r][col] = hs;
        }
    }
    __syncthreads();

    // lanes 0-15: nb rows 0..15;  lanes 16-31: self rows 0..15
    const int sel = khalf;
    const float* wa = Wa + sel * DD;
    float dot = 0.f;
    #pragma unroll 8
    for (int c = 0; c < DD; ++c) dot += lds[wave][sel][m][c] * wa[c];

    if (tile < n_tiles) {
        const int node = tile * 16 + m;
        if (sel == 0) a_nb[node]   = dot;
        else          a_self[node] = dot;
    }
}

// ---------------------------------------------------------------------------
// Edge pass: log_alpha -> hard-concrete gate -> mask (no atomics); L0 penalty
// accumulated with a wave-level reduction.
// ---------------------------------------------------------------------------
__global__ __launch_bounds__(256)
void edge_mask_kernel(const int* __restrict__ row, const int* __restrict__ col,
                      const float* __restrict__ u,
                      const float* __restrict__ a_nb, const float* __restrict__ a_self,
                      const float* __restrict__ ba, const float* __restrict__ temp,
                      float* __restrict__ mask, float* __restrict__ l0_acc, int E)
{
    const int e = blockIdx.x * blockDim.x + threadIdx.x;
    float l0p = 0.f;
    if (e < E) {
        const float t  = temp[0];
        const float la = a_nb[row[e]] + a_self[col[e]] + ba[0];
        const float uu = u[e];
        const float noise = __logf(uu) - log1pf(-uu);
        const float gate  = 1.f / (1.f + __expf(-(noise + la) / t));
        float mk = gate * (ZETA - GAMMA) + GAMMA;
        mk = fminf(fmaxf(mk, 0.f), 1.f);
        mask[e] = mk;
        l0p = 1.f / (1.f + __expf(-(la - t * LT_CONST)));
    }
    l0p = waveReduce(l0p);
    if ((threadIdx.x & 31) == 0) atomicAddF(l0_acc, l0p);
}

// ---------------- rowsum (CSR) fused with d^{-1/2}, atomic-free -------------
__global__ __launch_bounds__(256)
void rowsum_dinv_kernel(const int* __restrict__ off, const int* __restrict__ csr_eid,
                        const float* __restrict__ mask, float* __restrict__ dinv, int n)
{
    const int i = blockIdx.x * blockDim.x + threadIdx.x;
    if (i >= n) return;
    const int s = off[i], e = off[i + 1];
    float acc = 0.f;
    for (int p = s; p < e; ++p) acc += mask[csr_eid[p]];
    dinv[i] = fminf(rsqrtf(acc + 1e-6f), 10.0f);
}

// ---------------------------------------------------------------------------
// Atomic-free CSR SpMM: one wave per output row, 2 dims/lane in registers.
// x rows (38 MB) are L2-resident (192 MB L2) -> gathered reads run at L2 speed.
// Every row fully written -> no zero-fill of x1/x2 needed.
// ---------------------------------------------------------------------------
__global__ __launch_bounds__(256)
void spmm_csr_kernel(const int* __restrict__ off, const int* __restrict__ csr_col,
                     const int* __restrict__ csr_eid,
                     const float* __restrict__ mask, const float* __restrict__ dinv,
                     const float* __restrict__ x, float* __restrict__ xout, int n)
{
    const int wid  = (int)(((long)blockIdx.x * blockDim.x + threadIdx.x) >> 5);
    const int lane = threadIdx.x & 31;
    if (wid >= n) return;
    const int s = off[wid], e = off[wid + 1];
    const float dr = dinv[wid];
    float ax = 0.f, ay = 0.f;
    for (int p = s; p < e; ++p) {
        const int   c = csr_col[p];
        const float v = mask[csr_eid[p]] * dr * dinv[c];
        const float2 xv = ((const float2*)(x + (size_t)c * DD))[lane];
        ax += v * xv.x;
        ay += v * xv.y;
    }
    float2 o; o.x = ax; o.y = ay;
    ((float2*)(xout + (size_t)wid * DD))[lane] = o;
}

// ---------------------------------------------------------------------------
// BPR loss: one wave per triplet, xs = features + x1 + x2 formed on the fly.
// ---------------------------------------------------------------------------
__global__ __launch_bounds__(256)
void bpr_kernel(const float* __restrict__ feat, const float* __restrict__ x1,
                const float* __restrict__ x2,
                const int* __restrict__ users, const int* __restrict__ items,
                const int* __restrict__ negs, float* __restrict__ bpr_acc, int B)
{
    const int wid  = (int)(((long)blockIdx.x * blockDim.x + threadIdx.x) >> 5);
    const int lane = threadIdx.x & 31;
    if (wid >= B) return;
    const size_t ua = (size_t)users[wid] * DD;
    const size_t pa = (size_t)(N_USER + items[wid]) * DD;
    const size_t na = (size_t)(N_USER + negs[wid]) * DD;
    float ps = 0.f, ns = 0.f;
    #pragma unroll
    for (int i = 0; i < 2; ++i) {
        const int d = lane + 32 * i;
        const float av = feat[ua + d] + x1[ua + d] + x2[ua + d];
        const float pv = feat[pa + d] + x1[pa + d] + x2[pa + d];
        const float nv = feat[na + d] + x1[na + d] + x2[na + d];
        ps += av * pv;
        ns += av * nv;
    }
    ps = waveReduce(ps);
    ns = waveReduce(ns);
    if (lane == 0) {
        const float s  = ps - ns;
        const float sg = 1.f / (1.f + __expf(-s));
        atomicAddF(bpr_acc, -__logf(sg + 1e-5f));
    }
}

// ---------------- parameter L2 reg (tiny; one wave) ----------------
__global__ void reg_kernel(const float* p0, const float* p1, const float* p2,
                           const float* p3, const float* p4, const float* p5,
                           const float* p6, const float* p7, const float* p8,
                           const float* p9, const float* p10, const float* p11,
                           float* __restrict__ reg_acc)
{
    const float* ps[12] = {p0,p1,p2,p3,p4,p5,p6,p7,p8,p9,p10,p11};
    const int    sz[12] = {4096,64,4096,64,128,1, 4096,64,4096,64,128,1};
    float s = 0.f;
    for (int i = 0; i < 12; ++i)
        for (int k = threadIdx.x; k < sz[i]; k += 32) {
            const float v = ps[i][k];
            s += v * v;
        }
    s = waveReduce(s);
    if (threadIdx.x == 0) *reg_acc = s;
}

// ---------------- final scalar combine ----------------
__global__ void finalize_kernel(const float* __restrict__ acc, float* __restrict__ out) {
    // acc: [0]=l0 layer0 sum, [1]=l0 layer1 sum, [2]=bpr sum, [3]=reg sum
    out[0] = acc[2] / (float)BB
           + acc[3] * REG_W
           + (acc[0] / (float)EE + acc[1] / (float)EE) * LAMBDA0;
}

// ===========================================================================
extern "C" void kernel_launch(void* const* d_in, const int* in_sizes, int n_in,
                              void* d_out, int out_size, void* d_ws, size_t ws_size,
                              hipStream_t stream)
{
    const float* features = (const float*)d_in[0];
    const int*   row      = (const int*)  d_in[1];
    const int*   col      = (const int*)  d_in[2];
    const int*   users    = (const int*)  d_in[3];
    const int*   items    = (const int*)  d_in[4];
    const int*   negs     = (const int*)  d_in[5];
    const float* u0       = (const float*)d_in[6];
    const float* u1       = (const float*)d_in[7];
    const float* temp     = (const float*)d_in[8];
    const float* W_nb0    = (const float*)d_in[9];
    const float* b_nb0    = (const float*)d_in[10];
    const float* W_self0  = (const float*)d_in[11];
    const float* b_self0  = (const float*)d_in[12];
    const float* W_att0   = (const float*)d_in[13];
    const float* b_att0   = (const float*)d_in[14];
    const float* W_nb1    = (const float*)d_in[15];
    const float* b_nb1    = (const float*)d_in[16];
    const float* W_self1  = (const float*)d_in[17];
    const float* b_self1  = (const float*)d_in[18];
    const float* W_att1   = (const float*)d_in[19];
    const float* b_att1   = (const float*)d_in[20];

    // ---- workspace layout; ~93 MB total ----
    const size_t ND = (size_t)NN * DD;
    float* fw     = (float*)d_ws;
    float* x1     = fw;  fw += ND;
    float* x2     = fw;  fw += ND;
    float* a_nb   = fw;  fw += NN;
    float* a_self = fw;  fw += NN;
    float* dinv   = fw;  fw += NN;
    float* mask   = fw;  fw += EE;
    float* acc    = fw;  fw += 8;
    int*  iw      = (int*)fw;
    int*  cnt     = iw;  iw += NN;
    int*  off     = iw;  iw += NN + 1;
    int*  cursor  = iw;  iw += NN;
    int*  csr_col = iw;  iw += EE;
    int*  csr_eid = iw;  iw += EE;

    const int n_tiles = NN / 16;                       // 9375 exactly
    const dim3 blk1024(1024), blk256(256), blk128(128), blk32(32), one(1);
    const int gNode  = (n_tiles + TPB_TILES - 1) / TPB_TILES;
    const int gEdge  = (EE + 255) / 256;
    const int gN     = (NN + 255) / 256;
    const int gNWave = (int)(((long)NN * 32 + 255) / 256);
    const int gBpr   = (BB * 32 + 255) / 256;

    // zero the accumulators + degree counts (graph replays must be deterministic)
    zero_f_kernel<<<1, blk32, 0, stream>>>(acc, 8);
    zero_i_kernel<<<gN, blk256, 0, stream>>>(cnt, NN);

    // ---------------- CSR build (shared by both layers) ----------------
    count_kernel<<<gEdge, blk256, 0, stream>>>(row, cnt, EE);
    scan_kernel<<<1, blk1024, 0, stream>>>(cnt, off, NN);
    copy_i_kernel<<<gN, blk256, 0, stream>>>(off, cursor, NN);
    scatter_kernel<<<gEdge, blk256, 0, stream>>>(row, col, cursor, csr_col, csr_eid, EE);

    // ---------------- layer 0 ----------------
    node_attn_kernel<<<gNode, blk128, 0, stream>>>(features, W_nb0, b_nb0,
                                                   W_self0, b_self0, W_att0,
                                                   n_tiles, a_nb, a_self);
    edge_mask_kernel<<<gEdge, blk256, 0, stream>>>(row, col, u0, a_nb, a_self,
                                                   b_att0, temp, mask, &acc[0], EE);
    rowsum_dinv_kernel<<<gN, blk256, 0, stream>>>(off, csr_eid, mask, dinv, NN);
    spmm_csr_kernel<<<gNWave, blk256, 0, stream>>>(off, csr_col, csr_eid, mask, dinv,
                                                   features, x1, NN);

    // ---------------- layer 1 (input x1) ----------------
    node_attn_kernel<<<gNode, blk128, 0, stream>>>(x1, W_nb1, b_nb1,
                                                   W_self1, b_self1, W_att1,
                                                   n_tiles, a_nb, a_self);
    edge_mask_kernel<<<gEdge, blk256, 0, stream>>>(row, col, u1, a_nb, a_self,
                                                   b_att1, temp, mask, &acc[1], EE);
    rowsum_dinv_kernel<<<gN, blk256, 0, stream>>>(off, csr_eid, mask, dinv, NN);
    spmm_csr_kernel<<<gNWave, blk256, 0, stream>>>(off, csr_col, csr_eid, mask, dinv,
                                                   x1, x2, NN);

    // ---------------- losses ----------------
    bpr_kernel<<<gBpr, blk256, 0, stream>>>(features, x1, x2, users, items, negs,
                                            &acc[2], BB);
    reg_kernel<<<1, blk32, 0, stream>>>(W_nb0, b_nb0, W_self0, b_self0, W_att0, b_att0,
                                        W_nb1, b_nb1, W_self1, b_self1, W_att1, b_att1,
                                        &acc[3]);
    finalize_kernel<<<1, one, 0, stream>>>(acc, (float*)d_out);
}